// GCAEncoder_7533372637720
// MI455X (gfx1250) — compile-verified
//
#include <hip/hip_runtime.h>

#define E_EDGES 4194304
#define PADE 6
#define EP (E_EDGES + PADE)
#define NNODES 262144

typedef __attribute__((ext_vector_type(2))) float v2f;
typedef __attribute__((ext_vector_type(8))) float v8f;

__device__ __forceinline__ void atomAddF(float* p, float v) {
  unsafeAtomicAdd(p, v);   // native global_atomic_add_f32 on gfx1250
}

// ---- edge list init: src/dst from edge_index, w=1 for real edges, 0 for pad ----
__global__ void init_edges_k(const int* __restrict__ ei, int* __restrict__ src,
                             int* __restrict__ dst, float* __restrict__ w) {
  int i = blockIdx.x * blockDim.x + threadIdx.x;
  if (i >= EP) return;
  if (i < E_EDGES) { src[i] = ei[i]; dst[i] = ei[E_EDGES + i]; w[i] = 1.0f; }
  else             { src[i] = 0;     dst[i] = 0;               w[i] = 0.0f; }
}

__global__ void zero_k(float* __restrict__ p, int n) {
  int i = blockIdx.x * blockDim.x + threadIdx.x;
  if (i < n) p[i] = 0.0f;
}

// ---- deg = segment_sum(w, dst) ----
__global__ void deg_k(const int* __restrict__ dst, const float* __restrict__ w,
                      float* __restrict__ deg) {
  int i = blockIdx.x * blockDim.x + threadIdx.x;
  if (i >= EP) return;
  float wv = w[i];
  if (wv != 0.0f) atomAddF(&deg[dst[i]], wv);
}

// ---- dinv = rsqrt(deg + 1)  (self loop) ----
__global__ void dinv_k(const float* __restrict__ deg, float* __restrict__ dinv, int n) {
  int i = blockIdx.x * blockDim.x + threadIdx.x;
  if (i < n) dinv[i] = rsqrtf(deg[i] + 1.0f);
}

// ---- H = X @ W via V_WMMA_F32_16X16X4_F32 (Fin fixed = 32, Fout = 32 or 16) ----
// One wave => 16 rows.  All launches use full tiles (EXEC all-1s for WMMA).
__global__ void dense_wmma_k(const float* __restrict__ X, const float* __restrict__ W,
                             float* __restrict__ H, int nrows, int Fout) {
  const int Fin = 32;
  int wave = threadIdx.x >> 5;          // 8 waves per block
  int lane = threadIdx.x & 31;
  int l    = lane & 15;
  int half = lane >> 4;
  int r0 = (blockIdx.x * 8 + wave) * 16;
  if (r0 >= nrows) return;              // wave-uniform (never taken with our grids)

  // A fragments: 16x4 f32 slabs.  lanes 0-15: K0,K1 ; lanes 16-31: K2,K3
  v2f a[8];
  const float* xr = X + (size_t)(r0 + l) * Fin;
#pragma unroll
  for (int kb = 0; kb < 8; ++kb) {
    a[kb] = *(const v2f*)(xr + kb * 4 + 2 * half);
  }

  for (int c0 = 0; c0 < Fout; c0 += 16) {
    v8f c = {};
#pragma unroll
    for (int kb = 0; kb < 8; ++kb) {
      int k = kb * 4 + 2 * half;
      v2f b;
      b.x = W[(size_t)k * Fout + c0 + l];
      b.y = W[(size_t)(k + 1) * Fout + c0 + l];
      c = __builtin_amdgcn_wmma_f32_16x16x4_f32(false, a[kb], false, b,
                                                (short)0, c, false, false);
    }
    // D layout: lanes 0-15 -> rows r, lanes 16-31 -> rows 8+r
    float* hp = H + (size_t)r0 * Fout + c0 + l;
#pragma unroll
    for (int r = 0; r < 8; ++r) {
      hp[(size_t)(r + 8 * half) * Fout] = c[r];
    }
  }
}

// ---- scatter: agg[dst] += dinv[src]*dinv[dst]*w * h[src]  (one lane per feature) ----
template <int F>
__global__ void scatter_k(const int* __restrict__ src, const int* __restrict__ dst,
                          const float* __restrict__ w, const float* __restrict__ dinv,
                          const float* __restrict__ h, float* __restrict__ agg) {
  long long t = (long long)blockIdx.x * blockDim.x + threadIdx.x;
  int e = (int)(t / F);
  int f = (int)(t % F);
  if (e >= EP) return;
  float wv = w[e];
  if (wv == 0.0f) return;
  int s = src[e], d = dst[e];
  float norm = dinv[s] * dinv[d] * wv;
  atomAddF(&agg[(size_t)d * F + f], norm * h[(size_t)s * F + f]);
}

// ---- out = relu(agg + dinv^2 * h + bias) ----
template <int F>
__global__ void finalize_k(const float* __restrict__ h, const float* __restrict__ dinv,
                           const float* __restrict__ bias, float* __restrict__ agg,
                           int total) {
  int i = blockIdx.x * blockDim.x + threadIdx.x;
  if (i >= total) return;
  int node = i / F;
  int f = i % F;
  float di = dinv[node];
  float v = agg[i] + di * di * h[i] + bias[f];
  agg[i] = v > 0.0f ? v : 0.0f;
}

// ---- pairwise max pool over node pairs ----
template <int F>
__global__ void pool_k(const float* __restrict__ h, float* __restrict__ xp, int total) {
  int i = blockIdx.x * blockDim.x + threadIdx.x;
  if (i >= total) return;
  int node = i / F;
  int f = i % F;
  float a = h[(size_t)(2 * node) * F + f];
  float b = h[(size_t)(2 * node) * F + F + f];
  xp[i] = fmaxf(a, b);
}

// ---- pooling edge update: mask edges with an endpoint >= m, set two pad slots ----
__global__ void edge_update_k(int* __restrict__ src, int* __restrict__ dst,
                              float* __restrict__ w, int m, int slot) {
  int i = blockIdx.x * blockDim.x + threadIdx.x;
  if (i >= EP) return;
  if (i == slot)          { src[i] = m - 1; dst[i] = 0;     w[i] = 1.0f; }
  else if (i == slot + 1) { src[i] = 0;     dst[i] = m - 1; w[i] = 1.0f; }
  else {
    if (!(src[i] < m && dst[i] < m)) w[i] = 0.0f;
  }
}

__global__ void copy_k(const int* __restrict__ s, int* __restrict__ d, int n) {
  int i = blockIdx.x * blockDim.x + threadIdx.x;
  if (i < n) d[i] = s[i];
}

extern "C" void kernel_launch(void* const* d_in, const int* in_sizes, int n_in,
                              void* d_out, int out_size, void* d_ws, size_t ws_size,
                              hipStream_t stream) {
  const float* x  = (const float*)d_in[0];
  const int*   ei = (const int*)d_in[1];
  const float* W1 = (const float*)d_in[2];
  const float* b1 = (const float*)d_in[3];
  const float* W2 = (const float*)d_in[4];
  const float* b2 = (const float*)d_in[5];
  const float* W3 = (const float*)d_in[6];
  const float* b3 = (const float*)d_in[7];
  float* out = (float*)d_out;
  char* ws = (char*)d_ws;

  size_t off = 0;
  auto alloc = [&](size_t bytes) -> char* {
    char* p = ws + off;
    off += (bytes + 255) & ~(size_t)255;
    return p;
  };

  int*   srcw   = (int*)  alloc((size_t)EP * 4);
  int*   dstw   = (int*)  alloc((size_t)EP * 4);
  float* ww     = (float*)alloc((size_t)EP * 4);
  float* deg    = (float*)alloc((size_t)NNODES * 4);
  float* dinv   = (float*)alloc((size_t)NNODES * 4);
  float* hbuf   = (float*)alloc((size_t)NNODES * 32 * 4);   // reused per layer
  float* aggbuf = (float*)alloc((size_t)NNODES * 32 * 4);   // reused per layer
  float* p1     = (float*)alloc((size_t)(NNODES / 2) * 32 * 4);
  float* p2     = (float*)alloc((size_t)(NNODES / 4) * 32 * 4);

  const int B = 256;
  auto nb = [](long long n) { return (unsigned)((n + 255) / 256); };

  init_edges_k<<<nb(EP), B, 0, stream>>>(ei, srcw, dstw, ww);

  // ---------------- layer 1: n=262144, 32->32 ----------------
  {
    const int n = NNODES, F = 32;
    zero_k<<<nb(n), B, 0, stream>>>(deg, n);
    deg_k<<<nb(EP), B, 0, stream>>>(dstw, ww, deg);
    dinv_k<<<nb(n), B, 0, stream>>>(deg, dinv, n);
    dense_wmma_k<<<n / 128, B, 0, stream>>>(x, W1, hbuf, n, F);
    zero_k<<<nb((long long)n * F), B, 0, stream>>>(aggbuf, n * F);
    scatter_k<32><<<nb((long long)EP * F), B, 0, stream>>>(srcw, dstw, ww, dinv, hbuf, aggbuf);
    finalize_k<32><<<nb((long long)n * F), B, 0, stream>>>(hbuf, dinv, b1, aggbuf, n * F);
    pool_k<32><<<nb((long long)(n / 2) * F), B, 0, stream>>>(aggbuf, p1, (n / 2) * F);
    edge_update_k<<<nb(EP), B, 0, stream>>>(srcw, dstw, ww, n / 2, E_EDGES);
  }

  // ---------------- layer 2: n=131072, 32->32 ----------------
  {
    const int n = NNODES / 2, F = 32;
    zero_k<<<nb(n), B, 0, stream>>>(deg, n);
    deg_k<<<nb(EP), B, 0, stream>>>(dstw, ww, deg);
    dinv_k<<<nb(n), B, 0, stream>>>(deg, dinv, n);
    dense_wmma_k<<<n / 128, B, 0, stream>>>(p1, W2, hbuf, n, F);
    zero_k<<<nb((long long)n * F), B, 0, stream>>>(aggbuf, n * F);
    scatter_k<32><<<nb((long long)EP * F), B, 0, stream>>>(srcw, dstw, ww, dinv, hbuf, aggbuf);
    finalize_k<32><<<nb((long long)n * F), B, 0, stream>>>(hbuf, dinv, b2, aggbuf, n * F);
    pool_k<32><<<nb((long long)(n / 2) * F), B, 0, stream>>>(aggbuf, p2, (n / 2) * F);
    edge_update_k<<<nb(EP), B, 0, stream>>>(srcw, dstw, ww, n / 2, E_EDGES + 2);
  }

  // ---------------- layer 3: n=65536, 32->16 ----------------
  const int latentElems = (NNODES / 8) * 16;   // 524288
  {
    const int n = NNODES / 4, F = 16;
    zero_k<<<nb(n), B, 0, stream>>>(deg, n);
    deg_k<<<nb(EP), B, 0, stream>>>(dstw, ww, deg);
    dinv_k<<<nb(n), B, 0, stream>>>(deg, dinv, n);
    dense_wmma_k<<<n / 128, B, 0, stream>>>(p2, W3, hbuf, n, F);
    zero_k<<<nb((long long)n * F), B, 0, stream>>>(aggbuf, n * F);
    scatter_k<16><<<nb((long long)EP * F), B, 0, stream>>>(srcw, dstw, ww, dinv, hbuf, aggbuf);
    finalize_k<16><<<nb((long long)n * F), B, 0, stream>>>(hbuf, dinv, b3, aggbuf, n * F);
    // final pool writes latent straight into d_out
    pool_k<16><<<nb((long long)(n / 2) * F), B, 0, stream>>>(aggbuf, out, latentElems);
    edge_update_k<<<nb(EP), B, 0, stream>>>(srcw, dstw, ww, n / 2, E_EDGES + 4);
  }

  // ---------------- emit edge arrays: stack([src,dst]) then w, flat after latent ----
  int* outi = (int*)out;
  copy_k<<<nb(EP), B, 0, stream>>>(srcw, outi + latentElems, EP);
  copy_k<<<nb(EP), B, 0, stream>>>(dstw, outi + latentElems + EP, EP);
  copy_k<<<nb(EP), B, 0, stream>>>((const int*)ww, outi + latentElems + 2 * EP, EP);
}